// GAT_17892833755184
// MI455X (gfx1250) — compile-verified
//
#include <hip/hip_runtime.h>
#include <math.h>

#define DIM 128
#define HID 16
#define HEADS 4
#define NEG_SLOPE 0.2f

typedef __attribute__((ext_vector_type(2))) float v2f;
typedef __attribute__((ext_vector_type(8))) float v8f;

// ---------------------------------------------------------------------------
// Float atomic max via monotonic integer mapping (works for mixed signs,
// init value must be -FLT_MAX-ish). Avoids dependence on f32 max atomic API.
// ---------------------------------------------------------------------------
__device__ __forceinline__ void atomicMaxFloat(float* addr, float v) {
    if (v >= 0.0f) {
        atomicMax((int*)addr, __float_as_int(v));
    } else {
        atomicMin((unsigned int*)addr, __float_as_uint(v));
    }
}

// ---------------------------------------------------------------------------
// Kernel 1: h1 = x @ W1 using V_WMMA_F32_16X16X4_F32.
// Block = 128 threads (4 waves). Each wave owns a 16x64 output stripe.
// W1 (128x64 f32 = 32KB) is staged in LDS once per block.
//
// A (16x4 f32) layout: lane<16 holds row M=lane, (K0,K1); lane>=16 holds
// row M=lane-16, (K2,K3)  -> contiguous float2 load from x.
// B (4x16 f32) layout mirrors A: v0/v1 hold (Kk, Kk+1) for this lane half,
// N = lane & 15 (+ 16*ntile).
// C/D (16x16 f32): vgpr i holds rows M=i (lanes 0-15) and M=i+8 (lanes 16-31),
// N = lane & 15.
// ---------------------------------------------------------------------------
__global__ void __launch_bounds__(128)
gemm1_wmma(const float* __restrict__ x, const float* __restrict__ W1,
           float* __restrict__ h1, int N) {
    __shared__ float Wl[DIM * 64];
    for (int i = threadIdx.x; i < DIM * 64; i += blockDim.x) Wl[i] = W1[i];
    __syncthreads();

    const int wave  = threadIdx.x >> 5;
    const int lane  = threadIdx.x & 31;
    const int Mtile = blockIdx.x * 4 + wave;
    const int Mtiles = (N + 15) >> 4;
    if (Mtile >= Mtiles) return;            // wave-uniform: EXEC stays all-1s

    const int r  = lane & 15;               // row (A) / col (B,C,D) within tile
    const int kh = lane >> 4;               // k-half select
    const int mbase = Mtile * 16;
    int row = mbase + r;
    if (row >= N) row = N - 1;              // clamp (N is a multiple of 16 here)
    const float* __restrict__ xr = x + (size_t)row * DIM;

    v8f acc0 = {}, acc1 = {}, acc2 = {}, acc3 = {};

    for (int kb = 0; kb < DIM; kb += 4) {
        const int k0 = kb + 2 * kh;
        v2f a;
        a.x = xr[k0];
        a.y = xr[k0 + 1];
        const float* w0 = &Wl[(size_t)k0 * 64];
        const float* w1 = &Wl[(size_t)(k0 + 1) * 64];
        v2f b;
        b.x = w0[r];       b.y = w1[r];
        acc0 = __builtin_amdgcn_wmma_f32_16x16x4_f32(false, a, false, b, (short)0, acc0, false, false);
        b.x = w0[16 + r];  b.y = w1[16 + r];
        acc1 = __builtin_amdgcn_wmma_f32_16x16x4_f32(false, a, false, b, (short)0, acc1, false, false);
        b.x = w0[32 + r];  b.y = w1[32 + r];
        acc2 = __builtin_amdgcn_wmma_f32_16x16x4_f32(false, a, false, b, (short)0, acc2, false, false);
        b.x = w0[48 + r];  b.y = w1[48 + r];
        acc3 = __builtin_amdgcn_wmma_f32_16x16x4_f32(false, a, false, b, (short)0, acc3, false, false);
    }

    float* outp = h1 + (size_t)mbase * 64;
#pragma unroll
    for (int i = 0; i < 8; ++i) {
        const int m = i + 8 * kh;
        if (mbase + m < N) {
            float* o = outp + (size_t)m * 64 + r;
            o[0]  = acc0[i];
            o[16] = acc1[i];
            o[32] = acc2[i];
            o[48] = acc3[i];
        }
    }
}

// ---------------------------------------------------------------------------
// Kernel 2: per (node, head): attention logits alpha_s/alpha_d for layer 1,
// plus init of m1 (=-inf), s1 (=0) and o1 accumulator (=0).
// ---------------------------------------------------------------------------
__global__ void __launch_bounds__(256)
node_prep1(const float* __restrict__ h1, const float* __restrict__ attS,
           const float* __restrict__ attD, float* __restrict__ as1,
           float* __restrict__ ad1, float* __restrict__ m1,
           float* __restrict__ s1, float* __restrict__ o1, int N) {
    const int tid = blockIdx.x * blockDim.x + threadIdx.x;
    if (tid >= N * HEADS) return;
    const int n  = tid >> 2;
    const int hd = tid & 3;
    const float* hp = h1 + (size_t)n * 64 + hd * HID;
    float ss = 0.f, dd = 0.f;
#pragma unroll
    for (int c = 0; c < HID; ++c) {
        const float v = hp[c];
        ss += v * attS[hd * HID + c];
        dd += v * attD[hd * HID + c];
    }
    as1[tid] = ss;
    ad1[tid] = dd;
    m1[tid]  = -3.0e38f;
    s1[tid]  = 0.f;
    float* op = o1 + (size_t)n * 64 + hd * HID;
#pragma unroll
    for (int c = 0; c < HID; ++c) op[c] = 0.f;
}

// ---------------------------------------------------------------------------
// Kernel 3: per (edge, head): segment max of leaky-relu logits (layer 1).
// Edges E..E+N-1 are implicit self loops.
// ---------------------------------------------------------------------------
__global__ void __launch_bounds__(256)
edge_max1(const int* __restrict__ ei, const float* __restrict__ as1,
          const float* __restrict__ ad1, float* __restrict__ m1, int N, int E) {
    const int tid = blockIdx.x * blockDim.x + threadIdx.x;
    const int Et = E + N;
    const int e  = tid >> 2;
    const int hd = tid & 3;
    if (e >= Et) return;
    int s, d;
    if (e < E) { s = ei[e]; d = ei[E + e]; } else { s = d = e - E; }
    float le = as1[s * HEADS + hd] + ad1[d * HEADS + hd];
    le = (le > 0.f) ? le : NEG_SLOPE * le;
    atomicMaxFloat(&m1[d * HEADS + hd], le);
}

// ---------------------------------------------------------------------------
// Kernel 4: per (edge, head): w = exp(le - m[dst]); s1[dst] += w;
// o1[dst] += w * h1[src]  (un-normalized; divide by s later).
// ---------------------------------------------------------------------------
__global__ void __launch_bounds__(256)
edge_acc1(const int* __restrict__ ei, const float* __restrict__ as1,
          const float* __restrict__ ad1, const float* __restrict__ m1,
          float* __restrict__ s1, const float* __restrict__ h1,
          float* __restrict__ o1, int N, int E) {
    const int tid = blockIdx.x * blockDim.x + threadIdx.x;
    const int Et = E + N;
    const int e  = tid >> 2;
    const int hd = tid & 3;
    if (e >= Et) return;
    int s, d;
    if (e < E) { s = ei[e]; d = ei[E + e]; } else { s = d = e - E; }
    float le = as1[s * HEADS + hd] + ad1[d * HEADS + hd];
    le = (le > 0.f) ? le : NEG_SLOPE * le;
    const float w = __expf(le - m1[d * HEADS + hd]);
    atomicAdd(&s1[d * HEADS + hd], w);
    const float* hp = h1 + (size_t)s * 64 + hd * HID;
    float* op = o1 + (size_t)d * 64 + hd * HID;
#pragma unroll
    for (int c = 0; c < HID; ++c) atomicAdd(&op[c], w * hp[c]);
}

// ---------------------------------------------------------------------------
// Kernel 5: per element of o1: normalize by s1, add bias b1, apply ELU.
// Result stays in o1 (input to layer 2).
// ---------------------------------------------------------------------------
__global__ void __launch_bounds__(256)
finish1_elu(float* __restrict__ o1, const float* __restrict__ s1,
            const float* __restrict__ b1, int N) {
    const int i = blockIdx.x * blockDim.x + threadIdx.x;
    if (i >= N * 64) return;
    const int n  = i >> 6;
    const int hd = (i >> 4) & 3;
    float v = o1[i] / (s1[n * HEADS + hd] + 1e-16f) + b1[i & 63];
    v = (v > 0.f) ? v : expm1f(v);
    o1[i] = v;
}

// ---------------------------------------------------------------------------
// Kernel 6: per node: h2 = elu_h @ W2 (64->2), layer-2 attention logits,
// init m2/s2 and zero d_out accumulator.
// ---------------------------------------------------------------------------
__global__ void __launch_bounds__(256)
node_prep2(const float* __restrict__ o1, const float* __restrict__ W2,
           const float* __restrict__ attS2, const float* __restrict__ attD2,
           float* __restrict__ h2, float* __restrict__ as2,
           float* __restrict__ ad2, float* __restrict__ m2,
           float* __restrict__ s2, float* __restrict__ out, int N) {
    const int n = blockIdx.x * blockDim.x + threadIdx.x;
    if (n >= N) return;
    const float* hp = o1 + (size_t)n * 64;
    float a0 = 0.f, a1 = 0.f;
#pragma unroll
    for (int k = 0; k < 64; ++k) {
        const float v = hp[k];
        a0 += v * W2[k * 2 + 0];
        a1 += v * W2[k * 2 + 1];
    }
    h2[n * 2 + 0] = a0;
    h2[n * 2 + 1] = a1;
    as2[n] = a0 * attS2[0] + a1 * attS2[1];
    ad2[n] = a0 * attD2[0] + a1 * attD2[1];
    m2[n] = -3.0e38f;
    s2[n] = 0.f;
    out[n * 2 + 0] = 0.f;
    out[n * 2 + 1] = 0.f;
}

// ---------------------------------------------------------------------------
// Kernel 7: per edge: segment max (layer 2, single head).
// ---------------------------------------------------------------------------
__global__ void __launch_bounds__(256)
edge_max2(const int* __restrict__ ei, const float* __restrict__ as2,
          const float* __restrict__ ad2, float* __restrict__ m2, int N, int E) {
    const int e = blockIdx.x * blockDim.x + threadIdx.x;
    const int Et = E + N;
    if (e >= Et) return;
    int s, d;
    if (e < E) { s = ei[e]; d = ei[E + e]; } else { s = d = e - E; }
    float le = as2[s] + ad2[d];
    le = (le > 0.f) ? le : NEG_SLOPE * le;
    atomicMaxFloat(&m2[d], le);
}

// ---------------------------------------------------------------------------
// Kernel 8: per edge: accumulate un-normalized layer-2 messages into d_out.
// ---------------------------------------------------------------------------
__global__ void __launch_bounds__(256)
edge_acc2(const int* __restrict__ ei, const float* __restrict__ as2,
          const float* __restrict__ ad2, const float* __restrict__ m2,
          float* __restrict__ s2, const float* __restrict__ h2,
          float* __restrict__ out, int N, int E) {
    const int e = blockIdx.x * blockDim.x + threadIdx.x;
    const int Et = E + N;
    if (e >= Et) return;
    int s, d;
    if (e < E) { s = ei[e]; d = ei[E + e]; } else { s = d = e - E; }
    float le = as2[s] + ad2[d];
    le = (le > 0.f) ? le : NEG_SLOPE * le;
    const float w = __expf(le - m2[d]);
    atomicAdd(&s2[d], w);
    atomicAdd(&out[d * 2 + 0], w * h2[s * 2 + 0]);
    atomicAdd(&out[d * 2 + 1], w * h2[s * 2 + 1]);
}

// ---------------------------------------------------------------------------
// Kernel 9: per node: final normalize + bias.
// ---------------------------------------------------------------------------
__global__ void __launch_bounds__(256)
finish2(float* __restrict__ out, const float* __restrict__ s2,
        const float* __restrict__ b2, int N) {
    const int n = blockIdx.x * blockDim.x + threadIdx.x;
    if (n >= N) return;
    const float inv = 1.0f / (s2[n] + 1e-16f);
    out[n * 2 + 0] = out[n * 2 + 0] * inv + b2[0];
    out[n * 2 + 1] = out[n * 2 + 1] * inv + b2[1];
}

// ---------------------------------------------------------------------------
extern "C" void kernel_launch(void* const* d_in, const int* in_sizes, int n_in,
                              void* d_out, int out_size, void* d_ws, size_t ws_size,
                              hipStream_t stream) {
    const float* x    = (const float*)d_in[0];
    const int*   ei   = (const int*)d_in[1];
    const float* W1   = (const float*)d_in[2];
    const float* aS1  = (const float*)d_in[3];
    const float* aD1  = (const float*)d_in[4];
    const float* b1   = (const float*)d_in[5];
    const float* W2   = (const float*)d_in[6];
    const float* aS2  = (const float*)d_in[7];
    const float* aD2  = (const float*)d_in[8];
    const float* b2   = (const float*)d_in[9];

    const int N = in_sizes[0] / DIM;
    const int E = in_sizes[1] / 2;
    float* out = (float*)d_out;

    // Workspace layout (floats): ~60 MB total.
    float* ws  = (float*)d_ws;
    float* h1  = ws;                         // N*64
    float* as1 = h1 + (size_t)N * 64;        // N*4
    float* ad1 = as1 + (size_t)N * HEADS;    // N*4
    float* m1  = ad1 + (size_t)N * HEADS;    // N*4
    float* s1  = m1 + (size_t)N * HEADS;     // N*4
    float* o1  = s1 + (size_t)N * HEADS;     // N*64
    float* h2  = o1 + (size_t)N * 64;        // N*2
    float* as2 = h2 + (size_t)N * 2;         // N
    float* ad2 = as2 + (size_t)N;            // N
    float* m2  = ad2 + (size_t)N;            // N
    float* s2  = m2 + (size_t)N;             // N

    const int Et = E + N;
    const int Mtiles = (N + 15) >> 4;

    // Layer 1 GEMM (WMMA)
    gemm1_wmma<<<(Mtiles + 3) / 4, 128, 0, stream>>>(x, W1, h1, N);
    // Layer 1 node prep (logits + init)
    node_prep1<<<(N * HEADS + 255) / 256, 256, 0, stream>>>(h1, aS1, aD1, as1, ad1, m1, s1, o1, N);
    // Layer 1 segment softmax + aggregation
    edge_max1<<<((size_t)Et * HEADS + 255) / 256, 256, 0, stream>>>(ei, as1, ad1, m1, N, E);
    edge_acc1<<<((size_t)Et * HEADS + 255) / 256, 256, 0, stream>>>(ei, as1, ad1, m1, s1, h1, o1, N, E);
    finish1_elu<<<((size_t)N * 64 + 255) / 256, 256, 0, stream>>>(o1, s1, b1, N);
    // Layer 2 transform + logits + init
    node_prep2<<<(N + 255) / 256, 256, 0, stream>>>(o1, W2, aS2, aD2, h2, as2, ad2, m2, s2, out, N);
    // Layer 2 segment softmax + aggregation
    edge_max2<<<(Et + 255) / 256, 256, 0, stream>>>(ei, as2, ad2, m2, N, E);
    edge_acc2<<<(Et + 255) / 256, 256, 0, stream>>>(ei, as2, ad2, m2, s2, h2, out, N, E);
    finish2<<<(N + 255) / 256, 256, 0, stream>>>(out, s2, b2, N);
}